// ProbProtoSegHead_7799660610230
// MI455X (gfx1250) — compile-verified
//
#include <hip/hip_runtime.h>
#include <hip/hip_bf16.h>
#include <math.h>
#include <stdint.h>

// Problem constants (from reference): N=262144 pixels, D=256, C=19, M=10 -> 190 cols.
#define D_DIM   256
#define NCOL    190
#define NCOLP   192          // padded to 12 tiles of 16
#define NTILES  12           // column tiles of 16
#define ROWS    64           // rows per block (4 row quarters of 16)
#define PAD     260          // LDS row stride in floats (1040B: 16B aligned, bank-friendly)
#define LAMDA   (1.0f / 256.0f)
#define KSTEPS  64           // 256 / 4

typedef __attribute__((ext_vector_type(2))) float v2f;
typedef __attribute__((ext_vector_type(4))) float v4f;
typedef __attribute__((ext_vector_type(8))) float v8f;

// CDNA5 async global->LDS copy (16B per lane), tracked by ASYNCcnt.
// vdst = per-lane LDS byte offset, vaddr = per-lane 64-bit global address.
__device__ __forceinline__ void async_copy_g2l_b128(uint32_t lds_off,
                                                    const void* gaddr) {
  asm volatile("global_load_async_to_lds_b128 %0, %1, off"
               :: "v"(lds_off), "v"(gaddr)
               : "memory");
}
__device__ __forceinline__ void wait_asynccnt0() {
  asm volatile("s_wait_asynccnt 0x0" ::: "memory");
}

// ---------------------------------------------------------------------------
// Prep 1: build WMMA-B-fragment-swizzled, pre-scaled B matrices in workspace.
//   Bd[k][col] = 2 * prototypes[col][k]            (col = c*10+m, flat [190][256])
//   Bc[k][col] = 2*lamda * sqrt(proto_var[col][k])
// Fragment layout for V_WMMA_F32_16X16X4_F32 B (4x16):
//   lane holds {B[k0][col], B[k0+1][col]}, k0 = 4*ks + 2*(lane>>4), col = ct*16 + (lane&15)
// Packed: Bpack[((ct*64 + ks)*32 + lane)*2 + j]
// ---------------------------------------------------------------------------
__global__ __launch_bounds__(256) void prep_b_kernel(
    const float* __restrict__ proto, const float* __restrict__ pvar,
    float* __restrict__ Bd, float* __restrict__ Bc) {
  int g    = blockIdx.x * 256 + threadIdx.x;     // 12*64*32 = 24576 threads
  int ct   = g >> 11;                             // / 2048
  int rem  = g & 2047;
  int ks   = rem >> 5;
  int lane = rem & 31;
  int half = lane >> 4;
  int col  = ct * 16 + (lane & 15);
  int k0   = ks * 4 + half * 2;

  float bd0 = 0.f, bd1 = 0.f, bc0 = 0.f, bc1 = 0.f;
  if (col < NCOL) {
    const float* pr = proto + (size_t)col * D_DIM;
    const float* pv = pvar  + (size_t)col * D_DIM;
    bd0 = 2.0f * pr[k0];
    bd1 = 2.0f * pr[k0 + 1];
    bc0 = 2.0f * LAMDA * sqrtf(pv[k0]);
    bc1 = 2.0f * LAMDA * sqrtf(pv[k0 + 1]);
  }
  int idx = ((ct * (KSTEPS * 32)) + ks * 32 + lane) * 2;
  Bd[idx] = bd0; Bd[idx + 1] = bd1;
  Bc[idx] = bc0; Bc[idx + 1] = bc1;
}

// Prep 2: spl[col] = lamda * sum_k proto_var[col][k]
__global__ __launch_bounds__(192) void prep_spl_kernel(
    const float* __restrict__ pvar, float* __restrict__ spl) {
  int col = threadIdx.x;
  if (col >= NCOLP) return;
  float s = 0.f;
  if (col < NCOL) {
    const float* pv = pvar + (size_t)col * D_DIM;
    for (int k = 0; k < D_DIM; ++k) s += pv[k];
  }
  spl[col] = LAMDA * s;
}

// ---------------------------------------------------------------------------
// Main: each block = 64 rows x 192 cols. 12 wave32s, wave w owns col tile w,
// with 4 row-quarter accumulators. Two K-passes share one LDS tile buffer and
// accumulate into the same v8f accumulators:  acc = 2*dot + 2*lamda*cross.
// out[n][col] = acc - 2 - lamda*sum(x_var[n]) - lamda*sum(proto_var[col]).
// ---------------------------------------------------------------------------
__global__ __launch_bounds__(384) void probproto_wmma_kernel(
    const float* __restrict__ x, const float* __restrict__ xvar,
    const float* __restrict__ Bd, const float* __restrict__ Bc,
    const float* __restrict__ spl, float* __restrict__ out) {

  __shared__ float Xs[ROWS * PAD];     // 66,560 B staged tile (X, then sqrt(x_var))
  __shared__ float part[ROWS * 4];     // row-sum partials
  __shared__ float sxl[ROWS];          // lamda * sum_k x_var[row][k]

  const int tid   = threadIdx.x;
  const int lane  = tid & 31;
  const int w     = tid >> 5;          // column tile 0..11
  const int m     = lane & 15;
  const int half  = lane >> 4;
  const int koff  = half * 2;          // A/B fragment k sub-offset
  const size_t row0 = (size_t)blockIdx.x * ROWS;
  const int col   = w * 16 + m;

  v8f c0 = {};                         // rows  0..15 of tile
  v8f c1 = {};                         // rows 16..31
  v8f c2 = {};                         // rows 32..47
  v8f c3 = {};                         // rows 48..63

  // ---- Pass 1: stage X tile via CDNA5 async global->LDS (no VGPR round-trip)
  for (int t = tid; t < ROWS * (D_DIM / 4); t += 384) {
    int r  = t >> 6;
    int kq = (t & 63) << 2;
    uint32_t loff = (uint32_t)(uintptr_t)(Xs + r * PAD + kq);
    async_copy_g2l_b128(loff, (const void*)(x + ((row0 + r) << 8) + kq));
  }
  wait_asynccnt0();
  __syncthreads();

  // ---- Pass 1 GEMM: acc += (2*P) applied to X ----
  {
    const float* bw = Bd + (size_t)w * (KSTEPS * 64) + lane * 2;
#pragma unroll 4
    for (int ks = 0; ks < KSTEPS; ++ks) {
      int kk = ks * 4 + koff;
      v2f a0 = *(const v2f*)(Xs + m * PAD + kk);
      v2f a1 = *(const v2f*)(Xs + (m + 16) * PAD + kk);
      v2f a2 = *(const v2f*)(Xs + (m + 32) * PAD + kk);
      v2f a3 = *(const v2f*)(Xs + (m + 48) * PAD + kk);
      v2f b  = *(const v2f*)(bw + ks * 64);
      c0 = __builtin_amdgcn_wmma_f32_16x16x4_f32(false, a0, false, b,
                                                 (short)0, c0, false, false);
      c1 = __builtin_amdgcn_wmma_f32_16x16x4_f32(false, a1, false, b,
                                                 (short)0, c1, false, false);
      c2 = __builtin_amdgcn_wmma_f32_16x16x4_f32(false, a2, false, b,
                                                 (short)0, c2, false, false);
      c3 = __builtin_amdgcn_wmma_f32_16x16x4_f32(false, a3, false, b,
                                                 (short)0, c3, false, false);
    }
  }
  __syncthreads();   // all waves done reading Xs before overwrite

  // ---- Pass 2: stage sqrt(x_var) tile (transform => VGPR round-trip) ----
  for (int t = tid; t < ROWS * (D_DIM / 4); t += 384) {
    int r  = t >> 6;
    int kq = (t & 63) << 2;
    v4f v = __builtin_nontemporal_load(
        (const v4f*)(xvar + ((row0 + r) << 8) + kq));
    v4f s;
    s.x = sqrtf(v.x); s.y = sqrtf(v.y); s.z = sqrtf(v.z); s.w = sqrtf(v.w);
    *(v4f*)(Xs + r * PAD + kq) = s;
  }
  __syncthreads();

  // Deterministic row-sum of x_var (= sum of squares of staged sqrt values):
  // 4 partials per row, then a fixed-order reduction. No float atomics.
  if (tid < 256) {
    int r = tid >> 2, g = tid & 3;
    const float* p = Xs + r * PAD + g * 64;
    float s = 0.f;
#pragma unroll 8
    for (int kk = 0; kk < 64; ++kk) { float u = p[kk]; s += u * u; }
    part[r * 4 + g] = s;
  }
  __syncthreads();
  if (tid < ROWS) {
    float s = 0.f;
#pragma unroll
    for (int g = 0; g < 4; ++g) s += part[tid * 4 + g];
    sxl[tid] = LAMDA * s;
  }
  __syncthreads();

  // ---- Pass 2 GEMM: acc += (2*lamda*sqrt(PV)) applied to sqrt(x_var) ----
  {
    const float* bw = Bc + (size_t)w * (KSTEPS * 64) + lane * 2;
#pragma unroll 4
    for (int ks = 0; ks < KSTEPS; ++ks) {
      int kk = ks * 4 + koff;
      v2f a0 = *(const v2f*)(Xs + m * PAD + kk);
      v2f a1 = *(const v2f*)(Xs + (m + 16) * PAD + kk);
      v2f a2 = *(const v2f*)(Xs + (m + 32) * PAD + kk);
      v2f a3 = *(const v2f*)(Xs + (m + 48) * PAD + kk);
      v2f b  = *(const v2f*)(bw + ks * 64);
      c0 = __builtin_amdgcn_wmma_f32_16x16x4_f32(false, a0, false, b,
                                                 (short)0, c0, false, false);
      c1 = __builtin_amdgcn_wmma_f32_16x16x4_f32(false, a1, false, b,
                                                 (short)0, c1, false, false);
      c2 = __builtin_amdgcn_wmma_f32_16x16x4_f32(false, a2, false, b,
                                                 (short)0, c2, false, false);
      c3 = __builtin_amdgcn_wmma_f32_16x16x4_f32(false, a3, false, b,
                                                 (short)0, c3, false, false);
    }
  }

  // ---- Epilogue: C/D layout VGPR i -> row i + 8*half, col = lane&15 ----
  if (col < NCOL) {
    float bias_col = spl[col];
#pragma unroll
    for (int i = 0; i < 8; ++i) {
      int rr = i + half * 8;
      float o0 = c0[i] - 2.0f - sxl[rr]      - bias_col;
      float o1 = c1[i] - 2.0f - sxl[rr + 16] - bias_col;
      float o2 = c2[i] - 2.0f - sxl[rr + 32] - bias_col;
      float o3 = c3[i] - 2.0f - sxl[rr + 48] - bias_col;
      __builtin_nontemporal_store(o0, out + (row0 + rr)      * NCOL + col);
      __builtin_nontemporal_store(o1, out + (row0 + 16 + rr) * NCOL + col);
      __builtin_nontemporal_store(o2, out + (row0 + 32 + rr) * NCOL + col);
      __builtin_nontemporal_store(o3, out + (row0 + 48 + rr) * NCOL + col);
    }
  }
}

// ---------------------------------------------------------------------------
extern "C" void kernel_launch(void* const* d_in, const int* in_sizes, int n_in,
                              void* d_out, int out_size, void* d_ws, size_t ws_size,
                              hipStream_t stream) {
  const float* x     = (const float*)d_in[0];   // [N,256]
  const float* xvar  = (const float*)d_in[1];   // [N,256]
  const float* proto = (const float*)d_in[2];   // [19,10,256] -> [190,256]
  const float* pvar  = (const float*)d_in[3];   // [190,256]
  float* out = (float*)d_out;

  const int N = in_sizes[0] / D_DIM;            // 262144

  // Workspace layout (floats): Bd | Bc | spl
  const size_t bsz = (size_t)NTILES * KSTEPS * 64;   // 49152 floats each
  float* Bd  = (float*)d_ws;
  float* Bc  = Bd + bsz;
  float* spl = Bc + bsz;

  prep_b_kernel<<<(NTILES * KSTEPS * 32) / 256, 256, 0, stream>>>(proto, pvar, Bd, Bc);
  prep_spl_kernel<<<1, NCOLP, 0, stream>>>(pvar, spl);
  probproto_wmma_kernel<<<N / ROWS, 384, 0, stream>>>(x, xvar, Bd, Bc, spl, out);
}